// MultiBoxLoss_31456340476069
// MI455X (gfx1250) — compile-verified
//
#include <hip/hip_runtime.h>
#include <hip/hip_bf16.h>

// ---------------- problem constants (match reference) ----------------
constexpr int B = 64;
constexpr int P = 32768;
constexpr int G = 64;
constexpr int BLKB = P / 256;          // 128 blocks of 256 threads per batch
constexpr float VAR0 = 0.1f;
constexpr float THRESH = 0.35f;
constexpr int NEG_POS = 7;

// ---------------- workspace layout ----------------
constexpr size_t OFF_BTO  = 0;                              // B*P float
constexpr size_t OFF_BTI  = OFF_BTO + (size_t)B * P * 4;    // B*P int
constexpr size_t OFF_LC   = OFF_BTI + (size_t)B * P * 4;    // B*P float
constexpr size_t OFF_BPK  = OFF_LC  + (size_t)B * P * 4;    // B*G u64
constexpr size_t OFF_ANY  = OFF_BPK + (size_t)B * G * 8;    // B int
constexpr size_t OFF_PART = OFF_ANY + 256;                  // B*BLKB*4 float (16B aligned)
constexpr size_t OFF_NEG  = OFF_PART + (size_t)B * BLKB * 4 * 4;   // B float

// ---------------- small helpers ----------------
__device__ __forceinline__ float sl1(float d) {
    float a = fabsf(d);
    return a < 1.0f ? 0.5f * d * d : a - 0.5f;
}

__device__ __forceinline__ float waveSum(float v) {
#pragma unroll
    for (int o = 16; o > 0; o >>= 1) v += __shfl_xor(v, o, 32);
    return v;
}

typedef float v2f __attribute__((ext_vector_type(2)));
typedef float v8f __attribute__((ext_vector_type(8)));
typedef unsigned int v4u __attribute__((ext_vector_type(4)));
typedef int v8i __attribute__((ext_vector_type(8)));
typedef int v4i __attribute__((ext_vector_type(4)));

// Exact fp32 cross-lane (wave32) sum through the matrix unit (v_wmma_f32_16x16x4_f32).
// A=(x,0)/lane, B=ones -> D[m][n] = x[m]+x[m+16]; total = lane0 row-sum + lane16 row-sum.
__device__ __forceinline__ float wmmaSum(float x) {
#if defined(__AMDGCN__) && __has_builtin(__builtin_amdgcn_wmma_f32_16x16x4_f32)
    v2f a; a[0] = x;    a[1] = 0.0f;
    v2f bo; bo[0] = 1.0f; bo[1] = 1.0f;
    v8f c = {0.f,0.f,0.f,0.f,0.f,0.f,0.f,0.f};
    v8f d = __builtin_amdgcn_wmma_f32_16x16x4_f32(false, a, false, bo, (short)0, c, false, false);
    float s = d[0] + d[1] + d[2] + d[3] + d[4] + d[5] + d[6] + d[7];
    return __shfl(s, 0, 32) + __shfl(s, 16, 32);
#else
    for (int o = 16; o > 0; o >>= 1) x += __shfl_xor(x, o, 32);
    return x;
#endif
}

// ---------------- TDM staging of the per-batch targets block (64 rows x 15 f32 = 3840B) ----
#if defined(__AMDGCN__) && (__clang_major__ < 23) && \
    __has_builtin(__builtin_amdgcn_tensor_load_to_lds) && \
    __has_builtin(__builtin_amdgcn_s_wait_tensorcnt)
#define USE_TDM 1
#else
#define USE_TDM 0
#endif

// Loads G*15 contiguous floats from gsrc into LDS sdst; ends with a full barrier.
__device__ __forceinline__ void stage_targets(const float* __restrict__ gsrc,
                                              float* __restrict__ sdst, int tid) {
#if USE_TDM
    if (tid < 32) {  // wave 0 issues the DMA (TDM ignores EXEC; other waves branch around)
        const unsigned lds = (unsigned)(uintptr_t)sdst;
        const unsigned long long ga = (unsigned long long)(uintptr_t)gsrc;
        // D# group0: count=1, lds_addr, global_addr[56:0], type=2 (bits 127:126)
        v4u g0;
        g0[0] = 1u;
        g0[1] = lds;
        g0[2] = (unsigned)ga;
        g0[3] = (unsigned)(ga >> 32) | 0x80000000u;
        // D# group1: data_size=4B (2<<16); tensor_dim0=960, tensor_dim1=1,
        // tile_dim0=960, tile_dim1=1, tile_dim2=0, tensor_dim0_stride=960
        v8i g1;
        g1[0] = 0x00020000;              // mask=0 | data_size=2 | flags=0
        g1[1] = (int)(960u << 16);       // tensor_dim0[15:0] in [31:16]
        g1[2] = (int)(1u << 16);         // tensor_dim0[31:16]=0 | tensor_dim1[15:0]=1
        g1[3] = (int)(960u << 16);       // tensor_dim1[31:16]=0 | tile_dim0=960
        g1[4] = 1;                       // tile_dim1=1 | tile_dim2=0
        g1[5] = 960;                     // tensor_dim0_stride[31:0]
        g1[6] = 0;                       // stride0[47:32]=0 | stride1[15:0]=0
        g1[7] = 0;
        v4i z = {0, 0, 0, 0};            // groups 2/3 unused (2-D tensor)
        __builtin_amdgcn_tensor_load_to_lds(g0, g1, z, z, 0);
        __builtin_amdgcn_s_wait_tensorcnt((short)0);
    }
    __syncthreads();
#else
    for (int j = tid; j < G * 15; j += 256) sdst[j] = gsrc[j];
    __syncthreads();
#endif
}

// ---------------- kernel 1: bidirectional matching ----------------
__global__ __launch_bounds__(256) void k_match(const float* __restrict__ priors,
                                               const float* __restrict__ targets,
                                               float* __restrict__ bto,
                                               int* __restrict__ bti,
                                               unsigned long long* __restrict__ bpk) {
    const int b = blockIdx.y;
    const int tid = threadIdx.x;
    const int p = blockIdx.x * 256 + tid;

    __shared__ float sT[G * 15];               // full GT rows (boxes + landms + label)
    __shared__ float sArea[G];                 // precomputed GT areas
    __shared__ unsigned long long sKey[G];     // per-g block max key

    if (tid < G) sKey[tid] = 0ull;
    stage_targets(targets + (size_t)b * G * 15, sT, tid);   // includes barrier
    if (tid < G) {
        const float ax1 = sT[tid * 15 + 0], ay1 = sT[tid * 15 + 1];
        const float ax2 = sT[tid * 15 + 2], ay2 = sT[tid * 15 + 3];
        sArea[tid] = (ax2 - ax1) * (ay2 - ay1);
    }
    __syncthreads();

    const float4 pr = ((const float4*)priors)[p];
    const float bx1 = pr.x - 0.5f * pr.z, by1 = pr.y - 0.5f * pr.w;
    const float bx2 = pr.x + 0.5f * pr.z, by2 = pr.y + 0.5f * pr.w;
    const float areaB = pr.z * pr.w;

    float best = -1.0f; int bidx = 0;
#pragma unroll 4
    for (int g = 0; g < G; ++g) {
        const float ax1 = sT[g * 15 + 0], ay1 = sT[g * 15 + 1];
        const float ax2 = sT[g * 15 + 2], ay2 = sT[g * 15 + 3];
        float iw = fminf(ax2, bx2) - fmaxf(ax1, bx1); iw = fmaxf(iw, 0.0f);
        float ih = fminf(ay2, by2) - fmaxf(ay1, by1); ih = fmaxf(ih, 0.0f);
        const float inter = iw * ih;
        // 1-ulp reciprocal: single TRANS op, co-executes with VALU
        const float ov = inter * __builtin_amdgcn_rcpf(sArea[g] + areaB - inter);
        if (ov > best) { best = ov; bidx = g; }      // first-occurrence argmax over g
        // per-g argmax over priors: key = (bits(ov)<<32) | ~p  (ov>=0 -> monotone bits)
        // Guard on the hi word only (32-bit read, no tearing): skipping is safe iff
        // the current hi word is STRICTLY greater than ours; on <= we do the atomic.
        const unsigned hi = __float_as_uint(ov);
        volatile unsigned* cur = (volatile unsigned*)&sKey[g];
        if (cur[1] <= hi) {
            const unsigned long long key =
                ((unsigned long long)hi << 32) | (unsigned)(~(unsigned)p);
            atomicMax(&sKey[g], key);
        }
    }
    bto[(size_t)b * P + p] = best;
    bti[(size_t)b * P + p] = bidx;
    __syncthreads();
    if (tid < G) {
        const unsigned long long k = sKey[tid];
        volatile unsigned* cur = (volatile unsigned*)&bpk[b * G + tid];
        if (cur[1] <= (unsigned)(k >> 32)) atomicMax(&bpk[b * G + tid], k);
    }
}

// ---------------- kernel 2: apply best-prior overrides (sequential per batch) ----------------
__global__ void k_apply(const unsigned long long* __restrict__ bpk,
                        float* __restrict__ bto, int* __restrict__ bti,
                        int* __restrict__ anyv) {
    const int b = blockIdx.x;
    if (threadIdx.x != 0) return;
    int any = 0;
    for (int g = 0; g < G; ++g) {
        const unsigned long long key = bpk[b * G + g];
        const float ov = __uint_as_float((unsigned)(key >> 32));
        const unsigned pidx = ~(unsigned)key;
        const bool valid = ov >= 0.2f;
        any |= (int)valid;
        bti[(size_t)b * P + pidx] = g;  // unconditional scatter-set; last g wins
        if (valid) {
            float* d = &bto[(size_t)b * P + pidx];
            *d = fmaxf(*d, 2.0f);
        }
    }
    anyv[b] = any;
}

// ---------------- kernel 3: encode + per-prior losses + partials ----------------
__global__ __launch_bounds__(256) void k_loss(const float* __restrict__ loc,
                                              const float* __restrict__ conf,
                                              const float* __restrict__ landm,
                                              const float* __restrict__ priors,
                                              const float* __restrict__ targets,
                                              const float* __restrict__ bto,
                                              const int* __restrict__ bti,
                                              const int* __restrict__ anyv,
                                              float* __restrict__ lcneg,
                                              float* __restrict__ partials) {
    const int b = blockIdx.y, tid = threadIdx.x;
    const int p = blockIdx.x * 256 + tid;
    const size_t i = (size_t)b * P + p;

    __shared__ float sT[G * 15];
    stage_targets(targets + (size_t)b * G * 15, sT, tid);   // includes barrier

    const float ov = bto[i];
    const int g = bti[i];
    const float* tg = sT + g * 15;

    int ct = (int)tg[14];
    if (ov < THRESH) ct = 0;
    if (anyv[b] == 0) ct = 0;
    const bool pos = (ct != 0);

    const float4 pr = ((const float4*)priors)[p];
    const float cx = pr.x, cy = pr.y, pw = pr.z, ph = pr.w;
    const float ivw = 1.0f / (VAR0 * pw), ivh = 1.0f / (VAR0 * ph);

    float ll = 0.0f;
    if (pos) {
        const float x1 = tg[0], y1 = tg[1], x2 = tg[2], y2 = tg[3];
        const float gx = ((x1 + x2) * 0.5f - cx) * ivw;
        const float gy = ((y1 + y2) * 0.5f - cy) * ivh;
        const float gw = logf((x2 - x1) / pw) * 5.0f;   // 1/VAR1
        const float gh = logf((y2 - y1) / ph) * 5.0f;
        const float* ld = loc + i * 4;
        ll = sl1(ld[0] - gx) + sl1(ld[1] - gy) + sl1(ld[2] - gw) + sl1(ld[3] - gh);
    }

    float lm = 0.0f;
    if (ct == 1 || ct == 2) {
        const int kmax = (ct == 1) ? 5 : 2;   // face: all 10 dims; mask: first 4 dims
        const float* lp = landm + i * 10;
        for (int k = 0; k < kmax; ++k) {
            const float tx = (tg[4 + 2 * k] - cx) * ivw;
            const float ty = (tg[5 + 2 * k] - cy) * ivh;
            lm += sl1(lp[2 * k] - tx) + sl1(lp[2 * k + 1] - ty);
        }
    }

    const float* cd = conf + i * 3;
    const float c0 = cd[0], c1 = cd[1], c2 = cd[2];
    const float mx = fmaxf(c0, fmaxf(c1, c2));
    const float lse = mx + logf(expf(c0 - mx) + expf(c1 - mx) + expf(c2 - mx));
    const float gth = (ct == 0) ? c0 : ((ct == 1) ? c1 : c2);
    const float lc = lse - gth;

    lcneg[i] = pos ? 0.0f : lc;
    const float plc = pos ? lc : 0.0f;
    const float pcf = pos ? 1.0f : 0.0f;

    __shared__ float sred[8][4];
    const float r0 = waveSum(ll), r1 = waveSum(lm), r2 = waveSum(plc), r3 = waveSum(pcf);
    if ((tid & 31) == 0) {
        const int w = tid >> 5;
        sred[w][0] = r0; sred[w][1] = r1; sred[w][2] = r2; sred[w][3] = r3;
    }
    __syncthreads();
    if (tid == 0) {
        float t0 = 0, t1 = 0, t2 = 0, t3 = 0;
        for (int w = 0; w < 8; ++w) { t0 += sred[w][0]; t1 += sred[w][1]; t2 += sred[w][2]; t3 += sred[w][3]; }
        const size_t pb = ((size_t)b * BLKB + blockIdx.x) * 4;
        partials[pb + 0] = t0; partials[pb + 1] = t1; partials[pb + 2] = t2; partials[pb + 3] = t3;
    }
}

// ---------------- kernel 4: per-batch radix-select + negative sum ----------------
__global__ __launch_bounds__(256) void k_select(const float* __restrict__ lcneg,
                                                const float* __restrict__ partials,
                                                float* __restrict__ negsum) {
    const int b = blockIdx.x, tid = threadIdx.x;
    __shared__ unsigned int hist[256];
    __shared__ float sred[8];
    __shared__ unsigned int s_prefix;
    __shared__ int s_rem, s_numneg;

    float v = 0.0f;
    if (tid < BLKB) v = partials[((size_t)b * BLKB + tid) * 4 + 3];
    v = waveSum(v);
    if ((tid & 31) == 0) sred[tid >> 5] = v;
    __syncthreads();
    if (tid == 0) {
        float s = 0; for (int w = 0; w < 8; ++w) s += sred[w];
        int np = (int)(s + 0.5f);
        int nn = np * NEG_POS; if (nn > P - 1) nn = P - 1;
        s_numneg = nn; s_rem = nn; s_prefix = 0u;
    }
    __syncthreads();
    const int numneg = s_numneg;
    if (numneg <= 0) { if (tid == 0) negsum[b] = 0.0f; return; }

    const unsigned int* data = (const unsigned int*)(lcneg + (size_t)b * P);
    for (int pass = 3; pass >= 0; --pass) {
        const int shift = pass * 8;
        const unsigned int pm = (pass == 3) ? 0u : (0xFFFFFFFFu << (unsigned)((pass + 1) * 8));
        hist[tid] = 0u;
        __syncthreads();
        const unsigned int pref = s_prefix & pm;
        for (int idx = tid; idx < P; idx += 256) {
            const unsigned int u = data[idx];
            if ((u & pm) == pref) atomicAdd(&hist[(u >> shift) & 255], 1u);
        }
        __syncthreads();
        if (tid == 0) {
            int rem = s_rem, acc = 0, bin = 255;
            for (; bin >= 0; --bin) {
                const int c = (int)hist[bin];
                if (acc + c >= rem) break;
                acc += c;
            }
            if (bin < 0) bin = 0;
            s_rem = rem - acc;
            s_prefix = (s_prefix & pm) | ((unsigned)bin << shift);
        }
        __syncthreads();
    }
    const float Tf = __uint_as_float(s_prefix);
    const int r = s_rem;   // #ties at value T to include

    const float* lp = lcneg + (size_t)b * P;
    float s = 0.0f;
    for (int idx = tid; idx < P; idx += 256) {
        __builtin_prefetch(lp + idx + 4096, 0, 1);   // global_prefetch_b8 (CDNA5)
        const float x = lp[idx];
        s += (x > Tf) ? x : 0.0f;
    }
    s = waveSum(s);
    if ((tid & 31) == 0) sred[tid >> 5] = s;
    __syncthreads();
    if (tid == 0) {
        float t = 0; for (int w = 0; w < 8; ++w) t += sred[w];
        negsum[b] = t + (float)r * Tf;
    }
}

// ---------------- kernel 5: final reduce (single wave, WMMA cross-lane sums) ----------------
__global__ __launch_bounds__(32) void k_final(const float* __restrict__ partials,
                                              const float* __restrict__ negsum,
                                              float* __restrict__ out) {
    const int lane = threadIdx.x;
    float a_ll = 0, a_lm = 0, a_plc = 0, a_pc = 0, a_ng = 0;
    const int total = B * BLKB;
    for (int j = lane; j < total; j += 32) {
        const float4 q = ((const float4*)partials)[j];
        a_ll += q.x; a_lm += q.y; a_plc += q.z; a_pc += q.w;
    }
    for (int j = lane; j < B; j += 32) a_ng += negsum[j];

    const float S_ll  = wmmaSum(a_ll);
    const float S_lm  = wmmaSum(a_lm);
    const float S_plc = wmmaSum(a_plc);
    const float S_pc  = wmmaSum(a_pc);
    const float S_ng  = wmmaSum(a_ng);

    if (lane == 0) {
        const float N = fmaxf(S_pc, 1.0f);    // (conf_t>0) count == pos count here, so N1==N
        out[0] = S_ll / N;
        out[1] = (S_plc + S_ng) / N;
        out[2] = S_lm / N;
    }
}

// ---------------- host launcher ----------------
extern "C" void kernel_launch(void* const* d_in, const int* in_sizes, int n_in,
                              void* d_out, int out_size, void* d_ws, size_t ws_size,
                              hipStream_t stream) {
    const float* loc     = (const float*)d_in[0];
    const float* conf    = (const float*)d_in[1];
    const float* landm   = (const float*)d_in[2];
    const float* priors  = (const float*)d_in[3];
    const float* targets = (const float*)d_in[4];
    float* out = (float*)d_out;

    char* ws = (char*)d_ws;
    float* bto = (float*)(ws + OFF_BTO);
    int*   bti = (int*)(ws + OFF_BTI);
    float* lcn = (float*)(ws + OFF_LC);
    unsigned long long* bpk = (unsigned long long*)(ws + OFF_BPK);
    int*   anyv = (int*)(ws + OFF_ANY);
    float* part = (float*)(ws + OFF_PART);
    float* negs = (float*)(ws + OFF_NEG);

    // bpk is consumed via atomicMax -> must start at 0 every call
    hipMemsetAsync(bpk, 0, (size_t)B * G * sizeof(unsigned long long), stream);

    dim3 gridBP(BLKB, B);
    k_match <<<gridBP, 256, 0, stream>>>(priors, targets, bto, bti, bpk);
    k_apply <<<dim3(B), 32, 0, stream>>>(bpk, bto, bti, anyv);
    k_loss  <<<gridBP, 256, 0, stream>>>(loc, conf, landm, priors, targets, bto, bti, anyv, lcn, part);
    k_select<<<dim3(B), 256, 0, stream>>>(lcn, part, negs);
    k_final <<<1, 32, 0, stream>>>(part, negs, out);
}